// HashGrid_19688130084947
// MI455X (gfx1250) — compile-verified
//
#include <hip/hip_runtime.h>
#include <cmath>

// ---------------------------------------------------------------------------
// HashGrid encode (16 levels x 2 feat, 2^19-entry tables) + MLP 32->64->3, fp32.
// MI455X plan: 64MB hash table is L2-resident (192MB L2) -> gathers are L2 hits.
// MLP runs on V_WMMA_F32_16X16X4_F32 (native fp32 WMMA, exact-precision path),
// with the encoder producing A-fragments directly in the WMMA register layout.
// One wave = one 16-point tile: 8 levels/lane-half -> 8 A frags, 32 WMMAs.
// ---------------------------------------------------------------------------

typedef __attribute__((ext_vector_type(2))) float v2f;
typedef __attribute__((ext_vector_type(8))) float v8f;

#define TABLE_SIZE  (1u << 19)
#define HASH_MASK   (TABLE_SIZE - 1u)
#define PRIME1      2654435761u

struct Scales { float s[16]; };

__global__ __launch_bounds__(256) void hashgrid_mlp_kernel(
    const float2* __restrict__ points,   // [N] of (x,y)
    const float2* __restrict__ table,    // [16 * 524288] float2 features
    const float*  __restrict__ W1,       // [64][32] row-major
    const float*  __restrict__ W2,       // [3][64] row-major
    float*        __restrict__ out,      // [N][3]
    Scales sc, int npts)
{
    __shared__ float w2s[3 * 64];
    __shared__ float hbuf[8][16][68];    // per-wave 16x64 hidden, stride 68 (pad)

    const int tid   = threadIdx.x;
    const int wave  = tid >> 5;
    const int lane  = tid & 31;
    const int lhalf = lane >> 4;         // 0 or 1 (half-wave)
    const int lrow  = lane & 15;         // M index / N index within tile

    // Preload W2 (3x64) into LDS once per block.
    if (tid < 192) w2s[tid] = W2[tid];

    const int p0 = blockIdx.x * 128 + wave * 16;

    // --- W1 B-fragments: B[K][N] = W1[N][K].
    // Layout mirror of A: lane -> N = lrow, VGPR g -> K = 4*kk + g + 2*lhalf.
    v2f bfrag[4][8];
    {
        const int n = lrow;  // reused with +16*nb below
        #pragma unroll
        for (int nb = 0; nb < 4; ++nb) {
            #pragma unroll
            for (int kk = 0; kk < 8; ++kk) {
                const int row = 16 * nb + n;
                const int k   = 4 * kk + 2 * lhalf;   // even -> 8B aligned
                bfrag[nb][kk] = *(const v2f*)(W1 + row * 32 + k);
            }
        }
    }

    // --- Load this lane's point (both half-waves load the same 16 points).
    int pt = p0 + lrow;
    pt = (pt < npts) ? pt : (npts - 1);
    const float2 P = points[pt];

    // --- Hash-grid encode. Lane handles levels {2i + lhalf}, i = 0..7.
    // Result of iteration i is exactly WMMA A-fragment for k-step i:
    //   VGPR0 = enc[M][4i + 2*lhalf], VGPR1 = enc[M][4i + 2*lhalf + 1].
    v2f afrag[8];
    #pragma unroll
    for (int i = 0; i < 8; ++i) {
        const int   lvl   = 2 * i + lhalf;
        const float scale = sc.s[lvl];
        const float posx = P.x * scale;
        const float posy = P.y * scale;
        const float fx = floorf(posx), fy = floorf(posy);
        const float wx = posx - fx,    wy = posy - fy;
        const unsigned x0 = (unsigned)(int)fx;
        const unsigned y0 = (unsigned)(int)fy;
        const unsigned x1 = x0 + 1u,   y1 = y0 + 1u;
        const unsigned hy0 = y0 * PRIME1;
        const unsigned hy1 = y1 * PRIME1;
        const float2* tl = table + (size_t)lvl * TABLE_SIZE;
        // corners: (x0,y0) (x0,y1) (x1,y0) (x1,y1); PRIME0 == 1
        const float2 c00 = tl[(x0 ^ hy0) & HASH_MASK];
        const float2 c01 = tl[(x0 ^ hy1) & HASH_MASK];
        const float2 c10 = tl[(x1 ^ hy0) & HASH_MASK];
        const float2 c11 = tl[(x1 ^ hy1) & HASH_MASK];
        const float w00 = (1.0f - wx) * (1.0f - wy);
        const float w01 = (1.0f - wx) * wy;
        const float w10 = wx * (1.0f - wy);
        const float w11 = wx * wy;
        const float f0 = w00 * c00.x + w01 * c01.x + w10 * c10.x + w11 * c11.x;
        const float f1 = w00 * c00.y + w01 * c01.y + w10 * c10.y + w11 * c11.y;
        v2f av = { f0, f1 };
        afrag[i] = av;
    }

    // --- Layer 1 GEMM: (16x32) x (32x64) with fp32 WMMA, 8 k-steps x 4 n-blocks.
    v8f acc0 = {}, acc1 = {}, acc2 = {}, acc3 = {};
    #pragma unroll
    for (int kk = 0; kk < 8; ++kk) {
        acc0 = __builtin_amdgcn_wmma_f32_16x16x4_f32(false, afrag[kk], false, bfrag[0][kk], (short)0, acc0, false, false);
        acc1 = __builtin_amdgcn_wmma_f32_16x16x4_f32(false, afrag[kk], false, bfrag[1][kk], (short)0, acc1, false, false);
        acc2 = __builtin_amdgcn_wmma_f32_16x16x4_f32(false, afrag[kk], false, bfrag[2][kk], (short)0, acc2, false, false);
        acc3 = __builtin_amdgcn_wmma_f32_16x16x4_f32(false, afrag[kk], false, bfrag[3][kk], (short)0, acc3, false, false);
    }

    // --- ReLU + scatter C/D layout (lane l, vgpr v -> M = v + 8*lhalf, N = lrow)
    // into per-wave LDS tile (row-contiguous for layer 2).
    #pragma unroll
    for (int v = 0; v < 8; ++v) {
        const int m = v + 8 * lhalf;
        hbuf[wave][m][lrow +  0] = fmaxf(acc0[v], 0.0f);
        hbuf[wave][m][lrow + 16] = fmaxf(acc1[v], 0.0f);
        hbuf[wave][m][lrow + 32] = fmaxf(acc2[v], 0.0f);
        hbuf[wave][m][lrow + 48] = fmaxf(acc3[v], 0.0f);
    }

    __syncthreads();   // covers w2s preload + cross-lane hbuf visibility

    // --- Layer 2: rgb[m][j] = sum_n h[m][n] * W2[j][n]. 16 active lanes/wave.
    if (lane < 16) {
        const float* h = &hbuf[wave][lrow][0];
        float r0 = 0.0f, r1 = 0.0f, r2 = 0.0f;
        #pragma unroll
        for (int n = 0; n < 64; ++n) {
            const float hv = h[n];
            r0 = fmaf(hv, w2s[n],       r0);
            r1 = fmaf(hv, w2s[64 + n],  r1);
            r2 = fmaf(hv, w2s[128 + n], r2);
        }
        const int opt = p0 + lrow;
        if (opt < npts) {
            const int o = opt * 3;
            out[o + 0] = r0;
            out[o + 1] = r1;
            out[o + 2] = r2;
        }
    }
}

extern "C" void kernel_launch(void* const* d_in, const int* in_sizes, int n_in,
                              void* d_out, int out_size, void* d_ws, size_t ws_size,
                              hipStream_t stream) {
    (void)n_in; (void)d_ws; (void)ws_size; (void)out_size;
    const float2* points = (const float2*)d_in[0];
    const float2* table  = (const float2*)d_in[1];
    const float*  W1     = (const float*)d_in[2];
    const float*  W2     = (const float*)d_in[3];
    float* out = (float*)d_out;

    const int npts = in_sizes[0] / 2;

    // Replicate numpy: floor(16 * 1.447269237440378**l) in double, cast to f32.
    Scales sc;
    for (int l = 0; l < 16; ++l)
        sc.s[l] = (float)std::floor(16.0 * std::pow(1.447269237440378, (double)l));

    const int blocks = (npts + 127) / 128;   // 128 points per block (8 waves x 16)
    hashgrid_mlp_kernel<<<blocks, 256, 0, stream>>>(points, table, W1, W2, out, sc, npts);
}